// GCNEnc_19997367730789
// MI455X (gfx1250) — compile-verified
//
#include <hip/hip_runtime.h>
#include <hip/hip_bf16.h>

// ---------------------------------------------------------------------------
// GCN 2-layer encoder for MI455X (gfx1250, wave32).
// Dominant cost: edge scatter/gather over an L2-resident (25.6MB) feature
// matrix. GEMMs (50000x128 @ 128x128, fp32) use V_WMMA_F32_16X16X4_F32.
// ---------------------------------------------------------------------------

typedef __attribute__((ext_vector_type(2))) float v2f;
typedef __attribute__((ext_vector_type(8))) float v8f;

#define HID 128

// ---------------- elementwise / setup kernels ----------------

__global__ void zero_f32_kernel(float* __restrict__ p, int n) {
  int i = blockIdx.x * blockDim.x + threadIdx.x;
  if (i < n) p[i] = 0.0f;
}

__global__ void degree_kernel(const int* __restrict__ src, const int* __restrict__ dst,
                              float* __restrict__ deg_src, float* __restrict__ deg_dst,
                              int E) {
  int e = blockIdx.x * blockDim.x + threadIdx.x;
  if (e < E) {
    atomicAdd(&deg_src[src[e]], 1.0f);
    atomicAdd(&deg_dst[dst[e]], 1.0f);
  }
}

__global__ void rsqrt_kernel(float* __restrict__ p, int n) {
  int i = blockIdx.x * blockDim.x + threadIdx.x;
  if (i < n) {
    float v = p[i];
    p[i] = (v > 0.0f) ? rsqrtf(v) : 0.0f;  // zero-degree -> 0 (DGL convention)
  }
}

// ---------------- WMMA fp32 GEMM: out[M x 128] = diag(scale) * A[M x 128] @ W[128 x 128]
// One wave per 16-row M tile, full N=128 width (8 accumulator tiles).
// A 16x4 f32 layout (ISA 7.12.2): lanes 0-15 hold K=k0+0 (v0) / k0+1 (v1) for M=lane,
//                                 lanes 16-31 hold K=k0+2 / k0+3.
// B 4x16 f32 (mirror):            lanes 0-15 hold rows k0+0 / k0+1 at N=lane,
//                                 lanes 16-31 hold rows k0+2 / k0+3 at N=lane-16.
// D 16x16 f32: VGPR v -> M = v (lanes 0-15) / v+8 (lanes 16-31), N = lane&15.

__global__ void __launch_bounds__(128)
gemm_scaled_wmma(const float* __restrict__ A, const float* __restrict__ scale,
                 const float* __restrict__ W, float* __restrict__ out, int Mtiles) {
  const int lane  = threadIdx.x & 31;
  const int wave  = threadIdx.x >> 5;
  const int mtile = blockIdx.x * 4 + wave;
  if (mtile >= Mtiles) return;  // wave-uniform: EXEC stays all-ones for WMMA

  const int half  = lane >> 4;       // 0: lanes 0-15, 1: lanes 16-31
  const int sub   = lane & 15;
  const int row   = mtile * 16 + sub;
  const int koff  = half * 2;        // A/B K sub-offset for this half-wave
  const float s   = scale[row];
  const float* __restrict__ arow = A + (size_t)row * HID;

  v8f acc[8];
#pragma unroll
  for (int t = 0; t < 8; ++t) acc[t] = (v8f){};

  for (int k0 = 0; k0 < HID; k0 += 4) {
    v2f a;
    a[0] = arow[k0 + koff + 0] * s;
    a[1] = arow[k0 + koff + 1] * s;
#pragma unroll
    for (int nt = 0; nt < 8; ++nt) {
      v2f b;
      b[0] = W[(size_t)(k0 + koff + 0) * HID + nt * 16 + sub];
      b[1] = W[(size_t)(k0 + koff + 1) * HID + nt * 16 + sub];
      // 8 args: (neg_a, A, neg_b, B, c_mod, C, reuse_a, reuse_b)
      acc[nt] = __builtin_amdgcn_wmma_f32_16x16x4_f32(
          false, a, false, b, (short)0, acc[nt], false, false);
    }
  }

  // Store D tiles
  const int dmbase = mtile * 16 + half * 8;
#pragma unroll
  for (int nt = 0; nt < 8; ++nt) {
#pragma unroll
    for (int v = 0; v < 8; ++v) {
      out[(size_t)(dmbase + v) * HID + nt * 16 + sub] = acc[nt][v];
    }
  }
}

// ---------------- edge aggregation: agg[dst] += hW[src]  (L2-resident atomics)
// 128 consecutive threads handle one edge row -> coalesced read + atomic row.

__global__ void scatter_kernel(const float* __restrict__ hw, const int* __restrict__ src,
                               const int* __restrict__ dst, float* __restrict__ agg, int E) {
  int t = blockIdx.x * blockDim.x + threadIdx.x;
  int e = t >> 7;
  if (e < E) {
    int c = t & (HID - 1);
    atomicAdd(&agg[(size_t)dst[e] * HID + c], hw[(size_t)src[e] * HID + c]);
  }
}

// ---------------- h = relu(agg * norm_in + b)

__global__ void bias_relu_kernel(const float* __restrict__ agg, const float* __restrict__ nin,
                                 const float* __restrict__ bias, float* __restrict__ out,
                                 int total) {
  int t = blockIdx.x * blockDim.x + threadIdx.x;
  if (t < total) {
    int i = t >> 7, c = t & (HID - 1);
    float v = agg[t] * nin[i] + bias[c];
    out[t] = v > 0.0f ? v : 0.0f;
  }
}

// ---------------- layer-2 activation fused with column-sum pooling

__global__ void __launch_bounds__(256)
relu_pool_kernel(const float* __restrict__ agg, const float* __restrict__ nin,
                 const float* __restrict__ bias, float* __restrict__ pool, int N) {
  int t = blockIdx.x * blockDim.x + threadIdx.x;   // 512 row-groups x 128 cols
  int c = t & (HID - 1);
  int r0 = t >> 7;
  float b = bias[c];
  float acc = 0.0f;
  for (int i = r0; i < N; i += 512) {
    float v = agg[(size_t)i * HID + c] * nin[i] + b;
    acc += (v > 0.0f) ? v : 0.0f;
  }
  atomicAdd(&pool[c], acc);
}

// ---------------- out[1x2] = (pool/N) @ Wr + br

__global__ void final_kernel(const float* __restrict__ pool, const float* __restrict__ Wr,
                             const float* __restrict__ br, float* __restrict__ out,
                             float invN) {
  int j = threadIdx.x;
  if (j < 2) {
    float acc = 0.0f;
    for (int k = 0; k < HID; ++k) acc += pool[k] * invN * Wr[k * 2 + j];
    out[j] = acc + br[j];
  }
}

// ---------------- host-side orchestration ----------------

static inline int cdiv_i(int a, int b) { return (a + b - 1) / b; }

extern "C" void kernel_launch(void* const* d_in, const int* in_sizes, int n_in,
                              void* d_out, int out_size, void* d_ws, size_t ws_size,
                              hipStream_t stream) {
  const float* x  = (const float*)d_in[0];
  const int*   src = (const int*)d_in[1];
  const int*   dst = (const int*)d_in[2];
  const float* W1 = (const float*)d_in[3];
  const float* b1 = (const float*)d_in[4];
  const float* W2 = (const float*)d_in[5];
  const float* b2 = (const float*)d_in[6];
  const float* Wr = (const float*)d_in[7];
  const float* br = (const float*)d_in[8];
  float* out = (float*)d_out;

  const int N = in_sizes[0] / HID;   // 50000
  const int E = in_sizes[1];         // 640000

  // workspace layout (floats)
  float* ws      = (float*)d_ws;
  float* norm_out = ws;                         // [N]  (also degree accumulator)
  float* norm_in  = ws + N;                     // [N]
  float* buf0     = ws + 2 * (size_t)N;         // [N*HID]
  float* buf1     = buf0 + (size_t)N * HID;     // [N*HID]
  float* pool     = buf1 + (size_t)N * HID;     // [HID]

  const int T = 256;

  // degree norms (zero accumulators each call -> deterministic replays)
  zero_f32_kernel<<<cdiv_i(2 * N, T), T, 0, stream>>>(norm_out, 2 * N);
  zero_f32_kernel<<<1, T, 0, stream>>>(pool, HID);
  degree_kernel<<<cdiv_i(E, T), T, 0, stream>>>(src, dst, norm_out, norm_in, E);
  rsqrt_kernel<<<cdiv_i(2 * N, T), T, 0, stream>>>(norm_out, 2 * N);

  const int Mtiles = N / 16;                    // 3125 (exact for N=50000)
  dim3 gblocks(cdiv_i(Mtiles, 4));

  // layer 1: hW1 = diag(norm_out) x @ W1
  gemm_scaled_wmma<<<gblocks, 128, 0, stream>>>(x, norm_out, W1, buf0, Mtiles);
  zero_f32_kernel<<<cdiv_i(N * HID, T), T, 0, stream>>>(buf1, N * HID);
  scatter_kernel<<<cdiv_i(E * HID, T), T, 0, stream>>>(buf0, src, dst, buf1, E);
  bias_relu_kernel<<<cdiv_i(N * HID, T), T, 0, stream>>>(buf1, norm_in, b1, buf0, N * HID);

  // layer 2: hW2 = diag(norm_out) h @ W2
  gemm_scaled_wmma<<<gblocks, 128, 0, stream>>>(buf0, norm_out, W2, buf1, Mtiles);
  zero_f32_kernel<<<cdiv_i(N * HID, T), T, 0, stream>>>(buf0, N * HID);
  scatter_kernel<<<cdiv_i(E * HID, T), T, 0, stream>>>(buf1, src, dst, buf0, E);

  // activation + mean pool + readout
  relu_pool_kernel<<<256, 256, 0, stream>>>(buf0, norm_in, b2, pool, N);
  final_kernel<<<1, 32, 0, stream>>>(pool, Wr, br, out, 1.0f / (float)N);
}